// GateLoopAttention_83631603187968
// MI455X (gfx1250) — compile-verified
//
#include <hip/hip_runtime.h>
#include <math.h>

typedef float v2f __attribute__((ext_vector_type(2)));
typedef float v8f __attribute__((ext_vector_type(8)));
typedef unsigned int v4u __attribute__((ext_vector_type(4)));
typedef unsigned int v8u __attribute__((ext_vector_type(8)));

#define SEQ 4096
#define DIMX 512
#define DIM3 1536
#define NH 16
#define HD 32
#define CHUNK 256
#define NCHUNK 16   // SEQ / CHUNK

#define KB 32                 // K-block staged per TDM transfer
#define NKB (DIMX / KB)       // 16 K-blocks
#define BPADW 72              // padded LDS row stride in dwords (64 + 8 pad)

// ---------------------------------------------------------------------------
// TDM issue: load a KB x 64 f32 tile of W (row-major, row stride DIM3) into
// LDS at byte offset lds_off, padded to BPADW dwords per row.
// D# per CDNA5 ISA ch.8 (group0: count/lds/global/type, group1: dims/strides).
// Must be executed by a single wave; tracked with TENSORcnt.
// ---------------------------------------------------------------------------
__device__ __forceinline__ void tdm_load_b_tile(const float* W, int k0, int n0,
                                                unsigned lds_off) {
    unsigned long long ga =
        (unsigned long long)(const char*)(W + (size_t)k0 * DIM3 + n0);
    v4u g0;
    g0[0] = 1u;                                   // count=1, user mode
    g0[1] = lds_off;                              // lds_addr (bytes)
    g0[2] = (unsigned)ga;                         // global_addr[31:0]
    g0[3] = (unsigned)(ga >> 32) | 0x80000000u;   // addr[56:32] | type=2<<30

    v8u g1;
    g1[0] = (2u << 16)      // data_size = 4B
          | (1u << 20)      // pad_enable
          | (5u << 22)      // pad_interval: 64 dwords
          | (7u << 25);     // pad_amount: 8 dwords
    g1[1] = ((unsigned)(DIM3 & 0xFFFF)) << 16;                    // tensor_dim0 lo16
    g1[2] = (((unsigned)(DIMX & 0xFFFF)) << 16) | ((DIM3 >> 16) & 0xFFFF); // dim1 lo | dim0 hi
    g1[3] = (64u << 16) | ((DIMX >> 16) & 0xFFFF);                // tile_dim0=64 | dim1 hi
    g1[4] = (unsigned)KB;                                         // tile_dim1=KB, tile_dim2=0
    g1[5] = (unsigned)DIM3;                                       // tensor_dim0_stride lo32
    g1[6] = 0u;
    g1[7] = 0u;

    asm volatile("tensor_load_to_lds %0, %1" : : "s"(g0), "s"(g1) : "memory");
}

// ---------------------------------------------------------------------------
// Kernel 1: qkv = x @ W_qkv  (4096x512 @ 512x1536, all f32)
// WMMA f32 16x16x4. Block = 256 threads (8 waves); tile 128(M) x 64(N).
// B operand staged through LDS by the Tensor Data Mover, double-buffered.
// Grid: (1536/64, 4096/128) = (24, 32).
// ---------------------------------------------------------------------------
__global__ void __launch_bounds__(256)
gl_qkv_gemm_kernel(const float* __restrict__ x, const float* __restrict__ W,
                   float* __restrict__ qkv) {
    __shared__ float bsm[2][KB * BPADW];

    const int lane  = threadIdx.x & 31;
    const int wave  = threadIdx.x >> 5;
    const int m0    = blockIdx.y * 128 + wave * 16;
    const int n0    = blockIdx.x * 64;
    const int mlo   = lane & 15;       // M index within tile (A) / N index (B,C)
    const int khalf = lane >> 4;       // which K pair this half-wave holds

    const unsigned lds_base =
        (unsigned)(unsigned long long)(__attribute__((address_space(3))) float*)&bsm[0][0];
    const unsigned buf_stride = (unsigned)(KB * BPADW * sizeof(float));

    const float* xrow = x + (size_t)(m0 + mlo) * DIMX;

    v8f acc0 = {}, acc1 = {}, acc2 = {}, acc3 = {};

    if (wave == 0)
        tdm_load_b_tile(W, 0, n0, lds_base);

    for (int kb = 0; kb < NKB; kb++) {
        if (wave == 0)
            __builtin_amdgcn_s_wait_tensorcnt(0);
        __syncthreads();
        if (wave == 0 && kb + 1 < NKB)
            tdm_load_b_tile(W, (kb + 1) * KB, n0, lds_base + ((kb + 1) & 1) * buf_stride);

        const float* bb = &bsm[kb & 1][0];
        const int kbase = kb * KB;

#pragma unroll
        for (int ks = 0; ks < KB; ks += 4) {
            const int ka = ks + khalf * 2;
            // A fragment: lane L<16 -> {A[M][k],A[M][k+1]}, L>=16 -> {A[M][k+2],A[M][k+3]}
            v2f a = *(const v2f*)(xrow + kbase + ka);

            const float* br = bb + ka * BPADW + mlo;
            v2f b0; b0.x = br[0];            b0.y = br[BPADW];
            v2f b1; b1.x = br[16];           b1.y = br[BPADW + 16];
            v2f b2; b2.x = br[32];           b2.y = br[BPADW + 32];
            v2f b3; b3.x = br[48];           b3.y = br[BPADW + 48];

            acc0 = __builtin_amdgcn_wmma_f32_16x16x4_f32(false, a, false, b0, (short)0, acc0, false, false);
            acc1 = __builtin_amdgcn_wmma_f32_16x16x4_f32(false, a, false, b1, (short)0, acc1, false, false);
            acc2 = __builtin_amdgcn_wmma_f32_16x16x4_f32(false, a, false, b2, (short)0, acc2, false, false);
            acc3 = __builtin_amdgcn_wmma_f32_16x16x4_f32(false, a, false, b3, (short)0, acc3, false, false);
        }
    }

    // C/D layout: VGPR j holds row m0 + j + khalf*8, col n0 + t*16 + (lane&15)
#pragma unroll
    for (int j = 0; j < 8; j++) {
        const int row = m0 + j + khalf * 8;
        float* orow = qkv + (size_t)row * DIM3 + n0 + mlo;
        orow[0]  = acc0[j];
        orow[16] = acc1[j];
        orow[32] = acc2[j];
        orow[48] = acc3[j];
    }
}

// ---------------------------------------------------------------------------
// Kernel 2: gates. One wave per token n; lane j computes (x[n] . W_a[:,j]) + b_a[j].
// Even/odd lane pairs form the complex a; gate = (a/|a|) * sigmoid(|a|).
// gates layout: [h][n][2] (r, i).
// ---------------------------------------------------------------------------
__global__ void __launch_bounds__(256)
gl_gates_kernel(const float* __restrict__ x, const float* __restrict__ W_a,
                const float* __restrict__ b_a, float* __restrict__ gates) {
    const int lane = threadIdx.x & 31;
    const int wv   = threadIdx.x >> 5;
    const int n    = blockIdx.x * 8 + wv;
    if (n >= SEQ) return;

    float s = b_a[lane];
    const float* xr = x + (size_t)n * DIMX;
#pragma unroll 8
    for (int k = 0; k < DIMX; k++)
        s += xr[k] * W_a[k * 32 + lane];

    const float partner = __shfl_xor(s, 1, 32);
    const float ar = (lane & 1) ? partner : s;
    const float ai = (lane & 1) ? s : partner;
    const float mag = sqrtf(ar * ar + ai * ai);
    const float sig = 1.0f / (1.0f + __expf(-mag));
    const float scale = (mag > 0.0f) ? (sig / mag) : 0.0f;
    const int h = lane >> 1;
    gates[((size_t)h * SEQ + n) * 2 + (lane & 1)] = ((lane & 1) ? ai : ar) * scale;
}

// ---------------------------------------------------------------------------
// Kernel 3: chunk-local complex scan. Block = (head h, chunk c), 1024 threads.
// wave = d_v, lane = d_k (wave32-exact). State element kv[d_k][d_v] per thread.
// Emits: local output into d_out, per-position cumulative gate products P,
// and the chunk-final state.
// ---------------------------------------------------------------------------
__global__ void __launch_bounds__(1024)
gl_scan_chunk_kernel(const float* __restrict__ qkv, const float* __restrict__ gates,
                     float* __restrict__ out, float* __restrict__ stateBuf,
                     float* __restrict__ Pbuf) {
    const int c  = blockIdx.x;
    const int h  = blockIdx.y;
    const int dv = threadIdx.x >> 5;
    const int dk = threadIdx.x & 31;

    float s_r = 0.0f, s_i = 0.0f;      // kv state element
    float pr = 1.0f, pi = 0.0f;        // cumulative gate product

    const int base_n = c * CHUNK;
    for (int t = 0; t < CHUNK; t++) {
        const int n = base_n + t;
        const float gr = gates[((size_t)h * SEQ + n) * 2 + 0];
        const float gi = gates[((size_t)h * SEQ + n) * 2 + 1];
        const float* row = qkv + (size_t)n * DIM3 + h * HD;
        const float qv = row[dk];
        const float kv = row[DIMX + dk];
        const float vv = row[2 * DIMX + dv];

        const float nr = gr * s_r - gi * s_i + kv * vv;
        const float ni = gr * s_i + gi * s_r;
        s_r = nr; s_i = ni;

        float p = qv * s_r;            // out_n = q . Re(kv_n)
        p += __shfl_xor(p, 16, 32);
        p += __shfl_xor(p, 8, 32);
        p += __shfl_xor(p, 4, 32);
        p += __shfl_xor(p, 2, 32);
        p += __shfl_xor(p, 1, 32);
        if (dk == 0)
            out[(size_t)n * (NH * HD) + h * HD + dv] = p;

        const float npr = gr * pr - gi * pi;
        const float npi = gr * pi + gi * pr;
        pr = npr; pi = npi;
        if (threadIdx.x == 0) {
            Pbuf[((size_t)h * SEQ + n) * 2 + 0] = pr;
            Pbuf[((size_t)h * SEQ + n) * 2 + 1] = pi;
        }
    }

    const size_t sidx = (((size_t)h * NCHUNK + c) * 1024 + dv * 32 + dk) * 2;
    stateBuf[sidx]     = s_r;
    stateBuf[sidx + 1] = s_i;
}

// ---------------------------------------------------------------------------
// Kernel 4: cross-chunk carry scan (16 steps). One block per head, 1024 threads.
// C_{c+1} = G_c * C_c + S_c ; carryBuf[c] stores the carry-IN for chunk c.
// ---------------------------------------------------------------------------
__global__ void __launch_bounds__(1024)
gl_carry_kernel(const float* __restrict__ stateBuf, const float* __restrict__ Pbuf,
                float* __restrict__ carryBuf) {
    const int h = blockIdx.x;
    const int tid = threadIdx.x;
    float cr = 0.0f, ci = 0.0f;
    for (int c = 0; c < NCHUNK; c++) {
        const size_t idx = (((size_t)h * NCHUNK + c) * 1024 + tid) * 2;
        carryBuf[idx]     = cr;
        carryBuf[idx + 1] = ci;
        const float Gr = Pbuf[((size_t)h * SEQ + c * CHUNK + CHUNK - 1) * 2 + 0];
        const float Gi = Pbuf[((size_t)h * SEQ + c * CHUNK + CHUNK - 1) * 2 + 1];
        const float sr = stateBuf[idx], si = stateBuf[idx + 1];
        const float nr = Gr * cr - Gi * ci + sr;
        const float ni = Gr * ci + Gi * cr + si;
        cr = nr; ci = ni;
    }
}

// ---------------------------------------------------------------------------
// Kernel 5: correction. One wave per (h, n), lane = d_v.
// out += Pr*(q . Cr[:,dv]) - Pi*(q . Ci[:,dv])    (kv_n = P_n*C + kv_local)
// ---------------------------------------------------------------------------
__global__ void __launch_bounds__(256)
gl_correction_kernel(const float* __restrict__ qkv, const float* __restrict__ Pbuf,
                     const float* __restrict__ carryBuf, float* __restrict__ out) {
    const int lane = threadIdx.x & 31;
    const int wv   = threadIdx.x >> 5;
    const int gid  = blockIdx.x * 8 + wv;      // [0, NH*SEQ)
    const int h = gid >> 12;                   // gid / 4096
    const int n = gid & (SEQ - 1);
    const int c = n / CHUNK;

    const float* C = carryBuf + ((size_t)h * NCHUNK + c) * 1024 * 2;
    const float* q = qkv + (size_t)n * DIM3 + h * HD;

    float u = 0.0f, w = 0.0f;
#pragma unroll
    for (int dk = 0; dk < HD; dk++) {
        const float qv = q[dk];
        u += qv * C[(lane * 32 + dk) * 2 + 0];  // Cr[dk][dv=lane]
        w += qv * C[(lane * 32 + dk) * 2 + 1];  // Ci[dk][dv=lane]
    }
    const float Pr = Pbuf[((size_t)h * SEQ + n) * 2 + 0];
    const float Pi = Pbuf[((size_t)h * SEQ + n) * 2 + 1];
    out[(size_t)n * (NH * HD) + h * HD + lane] += Pr * u - Pi * w;
}

// ---------------------------------------------------------------------------
extern "C" void kernel_launch(void* const* d_in, const int* in_sizes, int n_in,
                              void* d_out, int out_size, void* d_ws, size_t ws_size,
                              hipStream_t stream) {
    const float* x     = (const float*)d_in[0];
    const float* W_qkv = (const float*)d_in[1];
    const float* W_a   = (const float*)d_in[2];
    const float* b_a   = (const float*)d_in[3];
    float* out = (float*)d_out;

    // Workspace layout
    char* ws = (char*)d_ws;
    float* qkv      = (float*)ws;                         ws += (size_t)SEQ * DIM3 * 4;             // 25.2 MB
    float* gates    = (float*)ws;                         ws += (size_t)NH * SEQ * 2 * 4;           // 0.5 MB
    float* Pbuf     = (float*)ws;                         ws += (size_t)NH * SEQ * 2 * 4;           // 0.5 MB
    float* stateBuf = (float*)ws;                         ws += (size_t)NH * NCHUNK * 1024 * 2 * 4; // 2 MB
    float* carryBuf = (float*)ws;                         // 2 MB

    // 1) qkv GEMM (WMMA f32 16x16x4, TDM-staged B operand)
    gl_qkv_gemm_kernel<<<dim3(DIM3 / 64, SEQ / 128), 256, 0, stream>>>(x, W_qkv, qkv);

    // 2) complex gates
    gl_gates_kernel<<<SEQ / 8, 256, 0, stream>>>(x, W_a, b_a, gates);

    // 3) chunk-local scan (also writes local part of the output)
    gl_scan_chunk_kernel<<<dim3(NCHUNK, NH), 1024, 0, stream>>>(qkv, gates, out, stateBuf, Pbuf);

    // 4) cross-chunk carry scan
    gl_carry_kernel<<<NH, 1024, 0, stream>>>(stateBuf, Pbuf, carryBuf);

    // 5) carry correction into the output
    gl_correction_kernel<<<(NH * SEQ) / 8, 256, 0, stream>>>(qkv, Pbuf, carryBuf, out);
}